// RandomVoxelSampling_21165598835473
// MI455X (gfx1250) — compile-verified
//
#include <hip/hip_runtime.h>
#include <hip/hip_bf16.h>

// ---------------- problem constants (from reference) ----------------
#define GX 352
#define GY 400
#define GZ 20
#define NGV (GX * GY * GZ)       // 2,816,000 voxels = 1375 * 2048 exactly
#define SAMPLE_NUM 16384
#define NPTS 200000
#define BATCH 8

#define RMINX 0.0f
#define RMINY (-40.0f)
#define RMINZ (-3.0f)
#define VOX   0.2f

// ---------------- tiling ----------------
#define TPB 256                   // threads per block (8 wave32 waves)
#define VPT 8                     // voxels per thread (contiguous -> stable vid order)
#define VPB (TPB * VPT)           // 2048 voxels per block
#define NBLK (NGV / VPB)          // 1375 blocks (divides exactly)

// ============================================================
// Tensor Data Mover: 1-D contiguous tile, global -> LDS.
// D# layout per CDNA5 ISA 8.3/8.4:
//   group0: count=1 | lds_addr | global_addr[56:0] | type=2
//   group1: data_size=3 (8B units), tensor_dim0=tile_dim0=n8,
//           tensor_dim1=tile_dim1=1, stride0=n8
// n8 = size in 8-byte units. Issue from one wave; EXEC is ignored by TDM.
// ============================================================
typedef unsigned int v4u __attribute__((ext_vector_type(4)));
typedef int          v8i __attribute__((ext_vector_type(8)));
typedef int          v4i __attribute__((ext_vector_type(4)));

__device__ __forceinline__ void tdm_load_1d(unsigned lds_byte_addr,
                                            const void* gaddr,
                                            unsigned n8) {
    const unsigned long long ga = (unsigned long long)gaddr;
    v4u g0;
    g0.x = 1u;                                        // count=1, is_restore=0
    g0.y = lds_byte_addr;                             // lds_addr (bytes)
    g0.z = (unsigned)(ga & 0xFFFFFFFFu);              // global_addr[31:0]
    g0.w = (unsigned)((ga >> 32) & 0x01FFFFFFu)       // global_addr[56:32]
         | (2u << 30);                                // type=2 ("image")
    v8i g1;
    g1[0] = (3 << 16);                                // wg_mask=0, data_size=3 (8B)
    g1[1] = (int)((n8 & 0xFFFFu) << 16);              // tensor_dim0[15:0]
    g1[2] = (int)((n8 >> 16) & 0xFFFFu) | (1 << 16);  // tensor_dim0[31:16], tensor_dim1=1
    g1[3] = (int)((n8 & 0xFFFFu) << 16);              // tile_dim0
    g1[4] = 1;                                        // tile_dim1=1, tile_dim2=0
    g1[5] = (int)n8;                                  // tensor_dim0_stride[31:0]
    g1[6] = 0;
    g1[7] = 0;
    const v4i z4 = {0, 0, 0, 0};
#if defined(__clang_major__) && (__clang_major__ >= 23)
    const v8i z8 = {0, 0, 0, 0, 0, 0, 0, 0};
    __builtin_amdgcn_tensor_load_to_lds(g0, g1, z4, z4, z8, 0);
#else
    __builtin_amdgcn_tensor_load_to_lds(g0, g1, z4, z4, 0);
#endif
}

__device__ __forceinline__ unsigned lds_addr_of(const void* p) {
    // Generic pointers to LDS carry the byte offset in the low 32 bits.
    return (unsigned)(unsigned long long)p;
}

// ============================================================
// 1) zero the SoA accumulator (4*NGV floats == NGV float4-stores)
// ============================================================
__global__ void k_zero_acc(float4* __restrict__ accw) {
    const float4 z = make_float4(0.f, 0.f, 0.f, 0.f);
    for (int v = blockIdx.x * blockDim.x + threadIdx.x; v < NGV;
         v += gridDim.x * blockDim.x) {
        accw[v] = z;
    }
}

// ============================================================
// 2) scatter: one thread per point; 4 f32 atomics into L2-resident SoA
// ============================================================
__global__ void k_scatter(const float* __restrict__ xyz,
                          float* __restrict__ sx, float* __restrict__ sy,
                          float* __restrict__ sz, float* __restrict__ cw) {
    const int i = blockIdx.x * blockDim.x + threadIdx.x;
    if (i >= NPTS) return;

    const int pf = i + 4096;                       // gfx1250 global_prefetch_b8
    if (pf < NPTS) __builtin_prefetch(xyz + (size_t)3 * pf, 0, 0);

    const float x = xyz[3 * (size_t)i + 0];
    const float y = xyz[3 * (size_t)i + 1];
    const float z = xyz[3 * (size_t)i + 2];

    // match reference: floor((p - rmin) / voxel)
    const int cx = (int)floorf((x - RMINX) / VOX);
    const int cy = (int)floorf((y - RMINY) / VOX);
    const int cz = (int)floorf((z - RMINZ) / VOX);

    const bool in_range = (cx >= 0) & (cx < GX) &
                          (cy >= 0) & (cy < GY) &
                          (cz >= 0) & (cz < GZ);
    if (!in_range) return;

    const int vid = (cx * GY + cy) * GZ + cz;
    atomicAdd(sx + vid, x);
    atomicAdd(sy + vid, y);
    atomicAdd(sz + vid, z);
    atomicAdd(cw + vid, 1.0f);
}

// ============================================================
// 3) per-block occupied count; counts tile staged via TDM -> LDS
// ============================================================
__global__ void k_block_count(const float* __restrict__ cw,
                              int* __restrict__ blockCnt) {
    __shared__ float lc[VPB];     // 8 KB counts tile
    __shared__ int s[TPB];
    const int blk = blockIdx.x;
    const int t   = threadIdx.x;

    if (t == 0) {
        tdm_load_1d(lds_addr_of(lc), cw + (size_t)blk * VPB, VPB * 4 / 8);
        __builtin_amdgcn_s_wait_tensorcnt(0);
    }
    __syncthreads();

    int c = 0;
#pragma unroll
    for (int k = 0; k < VPT; ++k)
        c += (lc[t * VPT + k] > 0.0f) ? 1 : 0;

    s[t] = c;
    __syncthreads();
#pragma unroll
    for (int off = TPB / 2; off > 0; off >>= 1) {
        if (t < off) s[t] += s[t + off];
        __syncthreads();
    }
    if (t == 0) blockCnt[blk] = s[0];
}

// ============================================================
// 4) single-block exclusive scan of 1375 block counts (2 per thread)
// ============================================================
__global__ void k_scan(const int* __restrict__ blockCnt,
                       int* __restrict__ blockOff,
                       int* __restrict__ tot) {
    __shared__ int s[1024];
    const int t  = threadIdx.x;
    const int i0 = 2 * t, i1 = 2 * t + 1;
    const int e0 = (i0 < NBLK) ? blockCnt[i0] : 0;
    const int e1 = (i1 < NBLK) ? blockCnt[i1] : 0;
    const int own = e0 + e1;
    s[t] = own;
    __syncthreads();
#pragma unroll
    for (int off = 1; off < 1024; off <<= 1) {
        const int v = (t >= off) ? s[t - off] : 0;
        __syncthreads();
        s[t] += v;
        __syncthreads();
    }
    const int excl = s[t] - own;
    if (i0 < NBLK) blockOff[i0] = excl;
    if (i1 < NBLK) blockOff[i1] = excl + e0;
    if (t == 1023) tot[0] = s[t];
}

// ============================================================
// 5) emit: early-exit when the whole block ranks past SAMPLE_NUM;
//    otherwise TDM-stage 4 SoA tiles, LDS-scan for stable rank, write means
// ============================================================
__global__ void k_emit(const float* __restrict__ sx, const float* __restrict__ sy,
                       const float* __restrict__ sz, const float* __restrict__ cw,
                       const int* __restrict__ blockOff,
                       float* __restrict__ out) {
    __shared__ float lx[VPB], ly[VPB], lz[VPB], lc[VPB];   // 4 x 8 KB
    __shared__ int s[TPB];
    const int blk = blockIdx.x;
    const int t   = threadIdx.x;

    const int base_rank = blockOff[blk];
    if (base_rank >= SAMPLE_NUM) return;     // ~92% of blocks skip all traffic

    if (t == 0) {
        const size_t off = (size_t)blk * VPB;
        const unsigned n8 = VPB * 4 / 8;
        tdm_load_1d(lds_addr_of(lx), sx + off, n8);
        tdm_load_1d(lds_addr_of(ly), sy + off, n8);
        tdm_load_1d(lds_addr_of(lz), sz + off, n8);
        tdm_load_1d(lds_addr_of(lc), cw + off, n8);
        __builtin_amdgcn_s_wait_tensorcnt(0);
    }
    __syncthreads();

    int c = 0;
#pragma unroll
    for (int k = 0; k < VPT; ++k)
        c += (lc[t * VPT + k] > 0.0f) ? 1 : 0;

    s[t] = c;
    __syncthreads();
#pragma unroll
    for (int off = 1; off < TPB; off <<= 1) {
        const int v = (t >= off) ? s[t - off] : 0;
        __syncthreads();
        s[t] += v;
        __syncthreads();
    }
    int rank = base_rank + s[t] - c;         // exclusive rank of this thread's 1st occ voxel

#pragma unroll
    for (int k = 0; k < VPT; ++k) {
        const int li = t * VPT + k;
        const float cnt = lc[li];
        if (cnt > 0.0f) {
            if (rank < SAMPLE_NUM) {
                const float inv = 1.0f / cnt;
                out[3 * (size_t)rank + 0] = lx[li] * inv;
                out[3 * (size_t)rank + 1] = ly[li] * inv;
                out[3 * (size_t)rank + 2] = lz[li] * inv;
            }
            rank++;
        }
    }
}

// ============================================================
// 6) tail fill for n_occ < SAMPLE_NUM (statistically unreachable here)
// ============================================================
__global__ void k_fill(const int* __restrict__ tot, float* __restrict__ out) {
    const int n = min(tot[0], SAMPLE_NUM);
    const int t = threadIdx.x;
    if (n <= 0) {
        for (int i = t; i < SAMPLE_NUM * 3; i += blockDim.x) out[i] = 0.0f;
        return;
    }
    for (int i = n + t; i < SAMPLE_NUM; i += blockDim.x) {
        const int src = i % n;
        out[3 * (size_t)i + 0] = out[3 * (size_t)src + 0];
        out[3 * (size_t)i + 1] = out[3 * (size_t)src + 1];
        out[3 * (size_t)i + 2] = out[3 * (size_t)src + 2];
    }
}

// ============================================================
// launcher: sequential over batches, SoA accumulator reused (~45 MB ws)
// ============================================================
extern "C" void kernel_launch(void* const* d_in, const int* in_sizes, int n_in,
                              void* d_out, int out_size, void* d_ws, size_t ws_size,
                              hipStream_t stream) {
    const float* xyz = (const float*)d_in[0];   // (8, 200000, 3) f32
    float* out = (float*)d_out;                 // (8, 16384, 3) f32

    // workspace layout: SoA accumulator then scan scratch
    char* w = (char*)d_ws;
    float* sx = (float*)w;
    float* sy = sx + NGV;
    float* sz = sy + NGV;
    float* cw = sz + NGV;
    int* blockCnt = (int*)(cw + NGV);
    int* blockOff = blockCnt + NBLK;
    int* tot      = blockOff + NBLK;

    for (int b = 0; b < BATCH; ++b) {
        const float* xyz_b = xyz + (size_t)b * NPTS * 3;
        float* out_b = out + (size_t)b * SAMPLE_NUM * 3;

        k_zero_acc   <<<2048, TPB, 0, stream>>>((float4*)w);
        k_scatter    <<<(NPTS + TPB - 1) / TPB, TPB, 0, stream>>>(xyz_b, sx, sy, sz, cw);
        k_block_count<<<NBLK, TPB, 0, stream>>>(cw, blockCnt);
        k_scan       <<<1, 1024, 0, stream>>>(blockCnt, blockOff, tot);
        k_emit       <<<NBLK, TPB, 0, stream>>>(sx, sy, sz, cw, blockOff, out_b);
        k_fill       <<<1, TPB, 0, stream>>>(tot, out_b);
    }
}